// AKF_80865644249656
// MI455X (gfx1250) — compile-verified
//
#include <hip/hip_runtime.h>
#include <hip/hip_bf16.h>

typedef __attribute__((ext_vector_type(2))) float v2f;
typedef __attribute__((ext_vector_type(8))) float v8f;

#define SSTR 65   // 64 + 1 padding -> conflict-free column reads
#define L_SEQ 500
#define BS 512
#define NS 64
#define NI 32

// One 16x16 output tile of a 64x64 GEMM, K accumulated 4 at a time with
// V_WMMA_F32_16X16X4_F32.  TA: read X transposed (A[m][k]=X[k][m]);
// TB: read Y transposed (B[k][n]=Y[n][k]).  Layouts per CDNA5 ISA 7.12.2:
//   A 16x4 : lanes 0-15 hold K=0,1 ; lanes 16-31 hold K=2,3
//   B 4x16 : mirrored ; C/D : VGPR r -> M=r (lanes 0-15), M=r+8 (lanes 16-31)
template<int TA, int TB>
__device__ __forceinline__ v8f mm16(const float* __restrict__ X,
                                    const float* __restrict__ Y,
                                    int rb, int cb, int half, int l, v8f acc) {
#pragma unroll
  for (int kk = 0; kk < NS; kk += 4) {
    const int k0 = kk + 2 * half;
    v2f a, b;
    if (TA) { a.x = X[k0 * SSTR + rb + l];     a.y = X[(k0 + 1) * SSTR + rb + l]; }
    else    { a.x = X[(rb + l) * SSTR + k0];   a.y = X[(rb + l) * SSTR + k0 + 1]; }
    if (TB) { b.x = Y[(cb + l) * SSTR + k0];   b.y = Y[(cb + l) * SSTR + k0 + 1]; }
    else    { b.x = Y[k0 * SSTR + cb + l];     b.y = Y[(k0 + 1) * SSTR + cb + l]; }
    acc = __builtin_amdgcn_wmma_f32_16x16x4_f32(false, a, false, b,
                                                (short)0, acc, false, false);
  }
  return acc;
}

__device__ __forceinline__ void store_tile(float* __restrict__ dst,
                                           int rb, int cb, int half, int l, v8f c) {
#pragma unroll
  for (int r = 0; r < 8; ++r)
    dst[(rb + r + 8 * half) * SSTR + cb + l] = c[r];
}

__global__ __launch_bounds__(256) void kf_wmma_kernel(
    const float* __restrict__ u_seq,   // (500,512,32)
    const float* __restrict__ y_seq,   // (500,512,64)
    const float* __restrict__ mu0,     // (512,64)
    const float* __restrict__ sig0,    // (512,64)
    const float* __restrict__ logsig,  // (64)
    const float* __restrict__ Bm,      // (64,32)
    const float* __restrict__ Hm,      // (64,64)
    const float* __restrict__ logdel,  // (64)
    float* __restrict__ out_mu,        // (500,512,64)
    float* __restrict__ out_ls)        // (500,512,64)
{
  __shared__ float sH[NS * SSTR];   // H
  __shared__ float sC[NS * SSTR];   // cov (prior -> pred -> post, in place)
  __shared__ float sT[NS * SSTR];   // T = H @ cov_pred
  __shared__ float sS[NS * SSTR];   // S -> S^-1 (in-place Gauss-Jordan)
  __shared__ float sK[NS * SSTR];   // K = T^T @ S^-1
  __shared__ float sQ[NS], sR[NS], sMu[NS], sMp[NS], sV[NS];

  const int b    = blockIdx.x;
  const int tid  = threadIdx.x;
  const int wave = tid >> 5, lane = tid & 31;
  const int half = lane >> 4, l = lane & 15;
  const int t0 = wave * 2, t1 = wave * 2 + 1;          // 2 tiles per wave
  const int rb0 = (t0 >> 2) * 16, cb0 = (t0 & 3) * 16;
  const int rb1 = (t1 >> 2) * 16, cb1 = (t1 & 3) * 16;

  // ---- init: H to LDS, cov0 = diag(sigma0^2), Q/R diagonals, t=0 outputs ----
  for (int i = tid; i < NS * NS; i += 256) {
    int r = i >> 6, c = i & 63;
    sH[r * SSTR + c] = Hm[i];
    float s0 = sig0[b * NS + r];
    sC[r * SSTR + c] = (r == c) ? s0 * s0 : 0.f;
  }
  if (tid < NS) {
    float e = __expf(logsig[tid]); sQ[tid] = e * e;
    float d = __expf(logdel[tid]); sR[tid] = d * d;
    float m = mu0[b * NS + tid];
    sMu[tid] = m;
    out_mu[(size_t)b * NS + tid] = m;                     // t = 0
    out_ls[(size_t)b * NS + tid] = __logf(sig0[b * NS + tid]);
  }
  __syncthreads();

  for (int t = 1; t < L_SEQ; ++t) {
    const float* u = u_seq + ((size_t)(t - 1) * BS + b) * NI;
    const float* y = y_seq + ((size_t)t * BS + b) * NS;

    // A: cov_pred = cov + Q (diag);  mu_pred = mu + B @ u
    if (tid < NS) {
      sC[tid * SSTR + tid] += sQ[tid];
      float a = sMu[tid];
#pragma unroll
      for (int j = 0; j < NI; ++j) a += Bm[tid * NI + j] * u[j];
      sMp[tid] = a;
    }
    __syncthreads();

    // B: T = H @ cov_pred
    { v8f z0 = {}; store_tile(sT, rb0, cb0, half, l, mm16<0,0>(sH, sC, rb0, cb0, half, l, z0));
      v8f z1 = {}; store_tile(sT, rb1, cb1, half, l, mm16<0,0>(sH, sC, rb1, cb1, half, l, z1)); }
    __syncthreads();

    // C: S = T @ H^T + diag(R)
    { v8f z0 = {}; v8f c0 = mm16<0,1>(sT, sH, rb0, cb0, half, l, z0);
#pragma unroll
      for (int r = 0; r < 8; ++r) {
        int row = rb0 + r + 8 * half, col = cb0 + l;
        sS[row * SSTR + col] = c0[r] + ((row == col) ? sR[row] : 0.f);
      }
      v8f z1 = {}; v8f c1 = mm16<0,1>(sT, sH, rb1, cb1, half, l, z1);
#pragma unroll
      for (int r = 0; r < 8; ++r) {
        int row = rb1 + r + 8 * half, col = cb1 + l;
        sS[row * SSTR + col] = c1[r] + ((row == col) ? sR[row] : 0.f);
      }
    }
    __syncthreads();

    // D: invert S in place (Gauss-Jordan, no pivoting: S is SPD)
    {
      const int gi = tid >> 2;   // row 0..63
      const int gq = tid & 3;    // 16-col quarter
      for (int p = 0; p < NS; ++p) {
        if (tid < NS) {
          float ip = 1.0f / sS[p * SSTR + p];
          float v  = sS[p * SSTR + tid];
          sS[p * SSTR + tid] = (tid == p) ? ip : v * ip;
        }
        __syncthreads();
        float f = sS[gi * SSTR + p];
        __syncthreads();
        if (gi != p) {
#pragma unroll
          for (int jj = 0; jj < 16; ++jj) {
            int j = gq * 16 + jj;
            float rp  = sS[p * SSTR + j];
            float cur = sS[gi * SSTR + j];
            sS[gi * SSTR + j] = (j == p) ? (-f * rp) : (cur - f * rp);
          }
        }
        __syncthreads();
      }
    }

    // E: K = T^T @ Sinv   (cov symmetric => cov @ H^T = T^T)
    { v8f z0 = {}; store_tile(sK, rb0, cb0, half, l, mm16<1,0>(sT, sS, rb0, cb0, half, l, z0));
      v8f z1 = {}; store_tile(sK, rb1, cb1, half, l, mm16<1,0>(sT, sS, rb1, cb1, half, l, z1)); }
    __syncthreads();

    // F1: innov = y - H @ mu_pred
    if (tid < NS) {
      float a = y[tid];
      for (int k = 0; k < NS; ++k) a -= sH[tid * SSTR + k] * sMp[k];
      sV[tid] = a;
    }
    __syncthreads();

    // G: cov_post = cov_pred - K @ T   (each output element RMW by one lane)
    { v8f z0 = {}; v8f c0 = mm16<0,0>(sK, sT, rb0, cb0, half, l, z0);
#pragma unroll
      for (int r = 0; r < 8; ++r) {
        int row = rb0 + r + 8 * half, col = cb0 + l;
        sC[row * SSTR + col] -= c0[r];
      }
      v8f z1 = {}; v8f c1 = mm16<0,0>(sK, sT, rb1, cb1, half, l, z1);
#pragma unroll
      for (int r = 0; r < 8; ++r) {
        int row = rb1 + r + 8 * half, col = cb1 + l;
        sC[row * SSTR + col] -= c1[r];
      }
    }
    // F2: mu_post = mu_pred + K @ innov
    if (tid < NS) {
      float m = sMp[tid];
      for (int k = 0; k < NS; ++k) m += sK[tid * SSTR + k] * sV[k];
      sMu[tid] = m;
      out_mu[((size_t)t * BS + b) * NS + tid] = m;
    }
    __syncthreads();

    // output log(sqrt(diag(cov_post)))
    if (tid < NS)
      out_ls[((size_t)t * BS + b) * NS + tid] = 0.5f * __logf(sC[tid * SSTR + tid]);
    __syncthreads();
  }
}

extern "C" void kernel_launch(void* const* d_in, const int* in_sizes, int n_in,
                              void* d_out, int out_size, void* d_ws, size_t ws_size,
                              hipStream_t stream) {
  const float* u    = (const float*)d_in[0];  // external_input_seq (500,512,32)
  const float* y    = (const float*)d_in[1];  // observations_seq  (500,512,64)
  const float* mu0  = (const float*)d_in[2];  // initial_prior_mu  (512,64)
  const float* sig0 = (const float*)d_in[3];  // initial_prior_sigma (512,64)
  const float* ls   = (const float*)d_in[4];  // logsigma (64)
  const float* Bm   = (const float*)d_in[5];  // B (64,32)
  const float* Hm   = (const float*)d_in[6];  // H (64,64)
  const float* ld   = (const float*)d_in[7];  // logdelta (64)
  float* out_mu = (float*)d_out;                              // (500,512,64)
  float* out_ls = out_mu + (size_t)L_SEQ * BS * NS;           // (500,512,64)
  kf_wmma_kernel<<<BS, 256, 0, stream>>>(u, y, mu0, sig0, ls, Bm, Hm, ld,
                                         out_mu, out_ls);
}